// SpodNet_61692910239928
// MI455X (gfx1250) — compile-verified
//
#include <hip/hip_runtime.h>

// ===========================================================================
// SpodNet (P=100, K=1, GY=1) for MI455X / gfx1250, wave32.
//
// Algebra exploited (derived from the reference):
//  * Final off-diagonals of Theta are batch-independent:
//      Theta[i,j] = rand_cols[max(i,j), min(i,j)]   (i != j)
//  * step2's theta_12 vectors are columns of that same matrix T.
//  * Only Theta[c,c] = 1 + t_c^T inv_T11 t_c is batch dependent, via the
//    recursion on W = inv(Theta):
//      y = W t_c ; v = y - (y_c/w22) W[:,c]  (v_c == 0)
//      schur = t.y - y_c^2 / w22
//      W <- W - (1/w22) u u^T + v v^T ; then W[:,c]=W[c,:]=-v, W[c,c]=1
//
// Heavy FLOPs are fp32 rank-2/rank-4 updates of a 112x112 LDS-resident
// matrix -> V_WMMA_F32_16X16X4_F32 per 16x16 tile.  W0 = inv(Theta) via
// in-place block Gauss-Jordan (4-wide pivots; SPD, no pivoting needed).
// Compute-bound (~20 GFLOP fp32 vs 160MB @ 23.3TB/s); fp32 WMMA is the
// highest-throughput instruction that preserves the accuracy an inverse
// recursion needs (bf16/fp8 WMMA would be numerically wrong).
// ===========================================================================

#define P       100
#define NP      112          // padded dim = 7 * 16
#define NT      7            // 16x16 tiles per side
#define NTILES  49
#define THREADS 256
#define NWAVES  8

typedef float v2f __attribute__((ext_vector_type(2)));
typedef float v8f __attribute__((ext_vector_type(8)));

// ---- one WMMA rank-4 tile update: W_tile += Apan_tile(16x4) * Bpan_tile(4x16)
// maskKb >= 0: zero C elements whose global row/col lies in [maskKb, maskKb+4)
__device__ __forceinline__ void wmma_rank4_tile(float* __restrict__ Wl,
                                                const float* __restrict__ Apan,
                                                const float* __restrict__ Bpan,
                                                int ti, int tj, int lane,
                                                int maskKb) {
  const int n  = lane & 15;          // N index (C/D layout)
  const int mo = (lane >> 4) * 8;    // M offset for C/D VGPRs
  const int m  = lane & 15;          // M index (A layout)
  const int k2 = (lane >> 4) * 2;    // K base  (A/B layout: {0,1}|{2,3})
  const int r0 = ti * 16;
  const int c0 = tj * 16;

  v8f c;
#pragma unroll
  for (int r = 0; r < 8; ++r) {
    int row = r0 + mo + r;
    int col = c0 + n;
    float w = Wl[row * NP + col];
    if (maskKb >= 0) {
      bool z = (row >= maskKb && row < maskKb + 4) ||
               (col >= maskKb && col < maskKb + 4);
      w = z ? 0.0f : w;
    }
    c[r] = w;
  }
  v2f a, b;
  a[0] = Apan[(r0 + m) * 4 + k2];
  a[1] = Apan[(r0 + m) * 4 + k2 + 1];
  b[0] = Bpan[k2 * NP + c0 + n];
  b[1] = Bpan[(k2 + 1) * NP + c0 + n];

  c = __builtin_amdgcn_wmma_f32_16x16x4_f32(false, a, false, b, (short)0, c,
                                            false, false);
#pragma unroll
  for (int r = 0; r < 8; ++r) Wl[(r0 + mo + r) * NP + c0 + n] = c[r];
}

// ---- batch-independent off-diagonal matrix T (padded 112x112) into d_ws ----
__global__ void spodnet_build_T(const float* __restrict__ rc,
                                float* __restrict__ Tmat) {
  int e = blockIdx.x * blockDim.x + threadIdx.x;
  if (e >= NP * NP) return;
  int i = e / NP, j = e % NP;
  float v = 0.0f;
  if (i < P && j < P && i != j)
    v = (i < j) ? rc[j * (P - 1) + i] : rc[i * (P - 1) + j];
  Tmat[e] = v;
}

// ---- main kernel: one workgroup (8 waves) per batch matrix ----------------
__global__ __launch_bounds__(THREADS) void spodnet_main(
    const float* __restrict__ Theta, const float* __restrict__ Tmat,
    float* __restrict__ Out) {
  __shared__ float Wl[NP * NP];     // 50.2 KB: W (in-place inverse + recursion)
  __shared__ float Apan[NP * 4];    // 16x4-operand column panel
  __shared__ float Bpan[4 * NP];    // 4x16-operand row panel
  __shared__ float PinvS[16];       // 4x4 pivot inverse
  __shared__ float tcl[NP];         // current t_c
  __shared__ float yl[NP];          // y = W t_c
  __shared__ float ypart[2 * NP];   // split-matvec partials
  __shared__ float diagl[NP];       // output diagonal

  const int tid  = threadIdx.x;
  const int lane = tid & 31;
  // wave id, forced into an SGPR: makes tile loops scalar (branch via
  // s_cbranch, EXEC stays all-ones as WMMA requires).
  const int wv   = __builtin_amdgcn_readfirstlane(tid >> 5);
  const int b    = blockIdx.x;
  const float* Tb = Theta + (size_t)b * (P * P);

  // Load Theta, pad with identity (padded block inverts to identity)
  for (int e = tid; e < NP * NP; e += THREADS) {
    int i = e / NP, j = e % NP;
    Wl[e] = (i < P && j < P) ? Tb[i * P + j] : ((i == j) ? 1.0f : 0.0f);
  }
  __syncthreads();

  // ---- Phase 1: in-place block Gauss-Jordan inversion ---------------------
  for (int kb = 0; kb < NP; kb += 4) {
    if (tid == 0) {  // invert 4x4 pivot block
      float M[4][4];
      for (int i = 0; i < 4; ++i)
        for (int j = 0; j < 4; ++j) M[i][j] = Wl[(kb + i) * NP + kb + j];
      for (int k = 0; k < 4; ++k) {
        float ip = 1.0f / M[k][k];
        M[k][k] = 1.0f;
        for (int j = 0; j < 4; ++j) M[k][j] *= ip;
        for (int i = 0; i < 4; ++i)
          if (i != k) {
            float f = M[i][k];
            M[i][k] = 0.0f;
            for (int j = 0; j < 4; ++j) M[i][j] -= f * M[k][j];
          }
      }
      for (int i = 0; i < 4; ++i)
        for (int j = 0; j < 4; ++j) PinvS[i * 4 + j] = M[i][j];
    }
    // Apan[i] = -A[i, kb..kb+3]; pivot rows get +I (so D row = Bpan row)
    for (int e = tid; e < NP * 4; e += THREADS) {
      int i = e >> 2, q = e & 3;
      float val;
      if (i >= kb && i < kb + 4)
        val = ((i - kb) == q) ? 1.0f : 0.0f;
      else
        val = -Wl[i * NP + kb + q];
      Apan[e] = val;
    }
    __syncthreads();
    // Bpan = Pinv * A[kb..kb+3, :], pivot-block columns replaced by Pinv
    for (int e = tid; e < 4 * NP; e += THREADS) {
      int r = e / NP, j = e % NP;
      float s;
      if (j >= kb && j < kb + 4) {
        s = PinvS[r * 4 + (j - kb)];
      } else {
        s = 0.0f;
        for (int q = 0; q < 4; ++q) s += PinvS[r * 4 + q] * Wl[(kb + q) * NP + j];
      }
      Bpan[e] = s;
    }
    __syncthreads();
    // rank-4 trailing update, one WMMA per 16x16 tile (scalar loop bounds)
    for (int t = wv; t < NTILES; t += NWAVES)
      wmma_rank4_tile(Wl, Apan, Bpan, t / NT, t % NT, lane, kb);
    __syncthreads();
  }
  // Wl now holds W0 = inv(Theta_b).

  // ---- Phase 2: 100 sequential column steps of the W recursion ------------
  for (int c = 0; c < P; ++c) {
    if (tid < NP) tcl[tid] = Tmat[c * NP + tid];  // t_c (T symmetric, row c)
    __syncthreads();

    // y = W t_c : 2 threads per row (halved serial chain), combine via LDS
    if (tid < 2 * NP) {
      int row = tid & (NP - 1);        // NP = 112 is not pow2 -> use explicit
      row = (tid < NP) ? tid : tid - NP;
      int half = (tid < NP) ? 0 : 1;
      const float* wr = &Wl[row * NP + half * (NP / 2)];
      const float* tv = &tcl[half * (NP / 2)];
      float s = 0.0f;
      for (int j = 0; j < NP / 2; ++j) s += wr[j] * tv[j];
      ypart[half * NP + row] = s;
    }
    __syncthreads();
    if (tid < NP) yl[tid] = ypart[tid] + ypart[NP + tid];
    __syncthreads();

    // wave 0: ty = t . y via lane-parallel partials + shfl_xor tree
    if (wv == 0) {
      float s = 0.0f;
      for (int j = lane; j < NP; j += 32) s += tcl[j] * yl[j];
#pragma unroll
      for (int off = 16; off > 0; off >>= 1) s += __shfl_xor(s, off, 32);
      if (lane == 0) {
        float w22 = Wl[c * NP + c];
        float yc  = yl[c];
        diagl[c] = 1.0f + s - yc * yc / w22;  // GY + schur
      }
    }
    // stage rank-2 panels:  W += (-u/w22) u^T + v v^T   (K padded to 4)
    if (tid < NP) {
      float w22   = Wl[c * NP + c];
      float alpha = yl[c] / w22;
      float wic   = Wl[tid * NP + c];     // u[i]
      float vi    = yl[tid] - alpha * wic;
      Apan[tid * 4 + 0] = -wic / w22;
      Apan[tid * 4 + 1] = vi;
      Apan[tid * 4 + 2] = 0.0f;
      Apan[tid * 4 + 3] = 0.0f;
      Bpan[0 * NP + tid] = Wl[c * NP + tid];  // u^T (row c, symmetry)
      Bpan[1 * NP + tid] = vi;                // v^T
      Bpan[2 * NP + tid] = 0.0f;
      Bpan[3 * NP + tid] = 0.0f;
    }
    __syncthreads();

    // rank-2 (K=4 padded) WMMA update of all tiles
    for (int t = wv; t < NTILES; t += NWAVES)
      wmma_rank4_tile(Wl, Apan, Bpan, t / NT, t % NT, lane, -1);
    __syncthreads();

    // row/col c overwrite: W[:,c] = W[c,:] = -v (v_c==0), W[c,c] = 1/GY = 1
    if (tid < NP) {
      float vi = Bpan[1 * NP + tid];
      float w  = (tid == c) ? 1.0f : -vi;
      Wl[tid * NP + c] = w;
      Wl[c * NP + tid] = w;
    }
    __syncthreads();
  }

  // ---- Output: off-diag = T (batch independent), diag = computed ----------
  float* ob = Out + (size_t)b * (P * P);
  for (int e = tid; e < P * P; e += THREADS) {
    int i = e / P, j = e % P;
    ob[e] = (i == j) ? diagl[i] : Tmat[i * NP + j];
  }
}

// ===========================================================================
extern "C" void kernel_launch(void* const* d_in, const int* in_sizes, int n_in,
                              void* d_out, int out_size, void* d_ws,
                              size_t ws_size, hipStream_t stream) {
  const float* Theta = (const float*)d_in[0];  // (B,100,100) f32
  const float* rc    = (const float*)d_in[1];  // (1,100,99)  f32
  float* Out  = (float*)d_out;                 // (B,100,100) f32
  float* Tmat = (float*)d_ws;                  // 112*112 f32 scratch

  const int B = in_sizes[0] / (P * P);

  spodnet_build_T<<<(NP * NP + 255) / 256, 256, 0, stream>>>(rc, Tmat);
  spodnet_main<<<dim3(B), dim3(THREADS), 0, stream>>>(Theta, Tmat, Out);
}